// GCN_88089779241192
// MI455X (gfx1250) — compile-verified
//
#include <hip/hip_runtime.h>

typedef __attribute__((ext_vector_type(2))) float v2f;
typedef __attribute__((ext_vector_type(8))) float v8f;

constexpr int N_NODES = 50000;
constexpr int N_EDGES = 800000;
constexpr int F_IN    = 128;
constexpr int F_H     = 256;
constexpr int F_OUT   = 40;
constexpr int NB      = (N_NODES + 255) / 256;  // 196 scan blocks

// ---------------- degree ----------------
__global__ void k_deg_init(float* deg) {
  int i = blockIdx.x * 256 + threadIdx.x;
  if (i < N_NODES) deg[i] = 1.0f;  // self-loop
}

__global__ void k_deg_edges(const int* __restrict__ dst, float* deg) {
  int e = blockIdx.x * 256 + threadIdx.x;
  if (e < N_EDGES) atomicAdd(&deg[dst[e]], 1.0f);
}

// ---------------- CSR build: exclusive scan of in-degree ----------------
__global__ __launch_bounds__(256) void k_scan1(const float* __restrict__ deg,
                                               int* __restrict__ rowstart,
                                               int* __restrict__ blocksum) {
  __shared__ int sh[256];
  const int t = threadIdx.x;
  const int i = blockIdx.x * 256 + t;
  const int c = (i < N_NODES) ? (int)(deg[i] - 0.5f) : 0;  // deg-1 (exact small ints)
  sh[t] = c;
  __syncthreads();
  for (int off = 1; off < 256; off <<= 1) {  // Hillis-Steele inclusive scan
    int v = (t >= off) ? sh[t - off] : 0;
    __syncthreads();
    sh[t] += v;
    __syncthreads();
  }
  if (i < N_NODES) rowstart[i] = sh[t] - c;  // exclusive within block
  if (t == 255) blocksum[blockIdx.x] = sh[255];
}

__global__ __launch_bounds__(256) void k_scan2(int* __restrict__ blocksum) {
  __shared__ int sh[256];
  const int t = threadIdx.x;
  const int c = (t < NB) ? blocksum[t] : 0;
  sh[t] = c;
  __syncthreads();
  for (int off = 1; off < 256; off <<= 1) {
    int v = (t >= off) ? sh[t - off] : 0;
    __syncthreads();
    sh[t] += v;
    __syncthreads();
  }
  if (t < NB) blocksum[t] = sh[t] - c;  // exclusive block offsets
}

__global__ void k_scan3(int* __restrict__ rowstart, const int* __restrict__ blocksum,
                        int* __restrict__ cursor) {
  const int i = blockIdx.x * 256 + threadIdx.x;
  if (i < N_NODES) {
    int r = rowstart[i] + blocksum[blockIdx.x];
    rowstart[i] = r;
    cursor[i]   = r;
  }
  if (i == 0) rowstart[N_NODES] = N_EDGES;
}

__global__ void k_inv(float* deg) {
  int i = blockIdx.x * 256 + threadIdx.x;
  if (i < N_NODES) deg[i] = rsqrtf(deg[i]);
}

// counting-sort edges by dst; precompute coef = inv[src]*inv[dst]
__global__ void k_place(const int* __restrict__ src, const int* __restrict__ dst,
                        const float* __restrict__ inv, int* __restrict__ cursor,
                        int* __restrict__ cs, float* __restrict__ val) {
  const int e = blockIdx.x * 256 + threadIdx.x;
  if (e < N_EDGES) {
    const int s = src[e], d = dst[e];
    const int p = atomicAdd(&cursor[d], 1);
    cs[p]  = s;
    val[p] = inv[s] * inv[d];
  }
}

// ---------------- fp32 WMMA GEMM: Y[N,256] = (relu?)X[N,K] @ W[K,256] ----------------
// 128 threads = 4 waves; block tile 64 rows x 64 cols; wave tile 16 x 64.
// LDS B tile stored PRE-PAIRED: Wl[p][col] = (W[2p][col], W[2p+1][col]) so each
// lane's B fragment is a single aligned ds_load_b64 into an even VGPR pair.
template <int K, bool RELU>
__global__ __launch_bounds__(128) void k_gemm_wmma(const float* __restrict__ X,
                                                   const float* __restrict__ W,
                                                   float* __restrict__ Y) {
  __shared__ float2 Wl[(K / 2) * 64];
  const int tid   = threadIdx.x;
  const int nbase = blockIdx.y * 64;

  for (int idx = tid; idx < (K / 2) * 16; idx += 128) {
    const int p = idx >> 4, c4 = idx & 15;
    float4 r0 = *(const float4*)(W + (size_t)(2 * p)     * F_H + nbase + c4 * 4);
    float4 r1 = *(const float4*)(W + (size_t)(2 * p + 1) * F_H + nbase + c4 * 4);
    float2* dp = Wl + p * 64 + c4 * 4;
    dp[0] = make_float2(r0.x, r1.x);
    dp[1] = make_float2(r0.y, r1.y);
    dp[2] = make_float2(r0.z, r1.z);
    dp[3] = make_float2(r0.w, r1.w);
  }
  __syncthreads();

  const int wave   = tid >> 5;
  const int lane   = tid & 31;
  const int tile_m = blockIdx.x * 4 + wave;
  if (tile_m * 16 >= N_NODES) return;  // N % 16 == 0

  const int lh  = lane >> 4;  // 0 -> K rows {k,k+1}; 1 -> {k+2,k+3}
  const int ll  = lane & 15;
  const int row = tile_m * 16 + ll;
  const float* Xr = X + (size_t)row * K;

  v8f acc[4] = {};
  for (int k = 0; k < K; k += 4) {
    const int p = (k >> 1) + lh;
    float2 av = *(const float2*)(Xr + k + lh * 2);
    float ax = av.x, ay = av.y;
    if (RELU) { ax = fmaxf(ax, 0.0f); ay = fmaxf(ay, 0.0f); }
    v2f a; a.x = ax; a.y = ay;
#pragma unroll
    for (int t = 0; t < 4; ++t) {
      float2 bv = Wl[p * 64 + t * 16 + ll];
      v2f b; b.x = bv.x; b.y = bv.y;
      acc[t] = __builtin_amdgcn_wmma_f32_16x16x4_f32(
          false, a, false, b, (short)0, acc[t], false, false);
    }
  }

  float* Yb = Y + (size_t)(tile_m * 16) * F_H + nbase;
#pragma unroll
  for (int t = 0; t < 4; ++t)
#pragma unroll
    for (int j = 0; j < 8; ++j) {
      const int r = j + lh * 8;  // D layout: VGPR j -> rows j / j+8
      Yb[(size_t)r * F_H + t * 16 + ll] = acc[t][j];
    }
}

// ---------------- CSR gather: agg[i] = tmp[i]*inv[i]^2 + b + sum_e coef*tmp[src] ----------------
// one wave per destination node; lane holds 2 float4s of the 256-wide row; no atomics.
__global__ __launch_bounds__(256) void k_gather(const float* __restrict__ tmp,
                                                const float* __restrict__ inv,
                                                const int* __restrict__ rowstart,
                                                const int* __restrict__ cs,
                                                const float* __restrict__ val,
                                                const float* __restrict__ b,
                                                float* __restrict__ agg) {
  const int wave = threadIdx.x >> 5;
  const int lane = threadIdx.x & 31;
  const int i = blockIdx.x * 8 + wave;
  if (i >= N_NODES) return;

  float s2 = inv[i]; s2 *= s2;
  const int q0 = lane, q1 = lane + 32;
  const float4* t4 = (const float4*)(tmp + (size_t)i * F_H);
  float4 a0 = t4[q0], a1 = t4[q1];
  float4 c0 = ((const float4*)b)[q0], c1 = ((const float4*)b)[q1];
  float4 acc0, acc1;
  acc0.x = fmaf(a0.x, s2, c0.x); acc0.y = fmaf(a0.y, s2, c0.y);
  acc0.z = fmaf(a0.z, s2, c0.z); acc0.w = fmaf(a0.w, s2, c0.w);
  acc1.x = fmaf(a1.x, s2, c1.x); acc1.y = fmaf(a1.y, s2, c1.y);
  acc1.z = fmaf(a1.z, s2, c1.z); acc1.w = fmaf(a1.w, s2, c1.w);

  const int p0 = rowstart[i], p1 = rowstart[i + 1];
  for (int p = p0; p < p1; ++p) {
    const int   s = cs[p];
    const float c = val[p];
    const float4* s4 = (const float4*)(tmp + (size_t)s * F_H);
    float4 v0 = s4[q0], v1 = s4[q1];
    acc0.x = fmaf(v0.x, c, acc0.x); acc0.y = fmaf(v0.y, c, acc0.y);
    acc0.z = fmaf(v0.z, c, acc0.z); acc0.w = fmaf(v0.w, c, acc0.w);
    acc1.x = fmaf(v1.x, c, acc1.x); acc1.y = fmaf(v1.y, c, acc1.y);
    acc1.z = fmaf(v1.z, c, acc1.z); acc1.w = fmaf(v1.w, c, acc1.w);
  }

  float4* o4 = (float4*)(agg + (size_t)i * F_H);
  o4[q0] = acc0;
  o4[q1] = acc1;
}

// ---------------- mean pool + head ----------------
__global__ void k_zero256(float* g) { g[threadIdx.x] = 0.0f; }

__global__ __launch_bounds__(256) void k_colsum(const float* __restrict__ h,
                                                float* __restrict__ g) {
  const int r0 = blockIdx.x * 64;
  const int c  = threadIdx.x;
  const int rend = min(r0 + 64, N_NODES);
  float acc = 0.0f;
  for (int r = r0; r < rend; ++r) acc += h[(size_t)r * F_H + c];
  atomicAdd(&g[c], acc);
}

__global__ void k_head(const float* __restrict__ g, const float* __restrict__ Wh,
                       const float* __restrict__ bh, float* __restrict__ out) {
  const int o = threadIdx.x;
  if (o >= F_OUT) return;
  const float invN = 1.0f / (float)N_NODES;
  float acc = bh[o];
  for (int k = 0; k < F_H; ++k) acc = fmaf(g[k] * invN, Wh[k * F_OUT + o], acc);
  out[o] = acc;
}

extern "C" void kernel_launch(void* const* d_in, const int* in_sizes, int n_in,
                              void* d_out, int out_size, void* d_ws, size_t ws_size,
                              hipStream_t stream) {
  (void)in_sizes; (void)n_in; (void)out_size; (void)ws_size;
  const float* x  = (const float*)d_in[0];
  const int*   ei = (const int*)d_in[1];
  const float* W0 = (const float*)d_in[2];
  const float* b0 = (const float*)d_in[3];
  const float* W1 = (const float*)d_in[4];
  const float* b1 = (const float*)d_in[5];
  const float* W2 = (const float*)d_in[6];
  const float* b2 = (const float*)d_in[7];
  const float* Wh = (const float*)d_in[8];
  const float* bh = (const float*)d_in[9];
  float* out = (float*)d_out;
  const int* src = ei;
  const int* dst = ei + N_EDGES;

  float* ws   = (float*)d_ws;
  float* inv  = ws;                              // N (deg -> rsqrt in place)
  float* gsum = ws + 50176;                      // 256
  float* bufA = ws + 50432;                      // N*256 gemm out
  float* bufB = bufA + (size_t)N_NODES * F_H;    // N*256 aggregated
  int*  rowstart = (int*)(ws + 25650432);        // N+1 (padded 50176)
  int*  cursor   = rowstart + 50176;             // N   (padded 50176)
  int*  blocksum = cursor + 50176;               // NB  (padded 256)
  int*  cs       = blocksum + 256;               // E
  float* val     = (float*)(cs + N_EDGES);       // E

  // degree + CSR build (one time, reused by all 3 layers)
  k_deg_init <<<NB, 256, 0, stream>>>(inv);
  k_deg_edges<<<(N_EDGES + 255) / 256, 256, 0, stream>>>(dst, inv);
  k_scan1    <<<NB, 256, 0, stream>>>(inv, rowstart, blocksum);
  k_scan2    <<<1, 256, 0, stream>>>(blocksum);
  k_scan3    <<<NB, 256, 0, stream>>>(rowstart, blocksum, cursor);
  k_inv      <<<NB, 256, 0, stream>>>(inv);
  k_place    <<<(N_EDGES + 255) / 256, 256, 0, stream>>>(src, dst, inv, cursor, cs, val);

  const dim3 gg((3125 + 3) / 4, F_H / 64, 1);
  const int  ggat = (N_NODES + 7) / 8;  // gather: 8 waves/block, wave per node

  // layer 0 (relu folded into next GEMM's A-load)
  k_gemm_wmma<F_IN, false><<<gg, 128, 0, stream>>>(x, W0, bufA);
  k_gather<<<ggat, 256, 0, stream>>>(bufA, inv, rowstart, cs, val, b0, bufB);
  // layer 1
  k_gemm_wmma<F_H, true><<<gg, 128, 0, stream>>>(bufB, W1, bufA);
  k_gather<<<ggat, 256, 0, stream>>>(bufA, inv, rowstart, cs, val, b1, bufB);
  // layer 2
  k_gemm_wmma<F_H, true><<<gg, 128, 0, stream>>>(bufB, W2, bufA);
  k_gather<<<ggat, 256, 0, stream>>>(bufA, inv, rowstart, cs, val, b2, bufB);

  // global mean pool + head
  k_zero256<<<1, 256, 0, stream>>>(gsum);
  k_colsum <<<(N_NODES + 63) / 64, 256, 0, stream>>>(bufB, gsum);
  k_head   <<<1, 64, 0, stream>>>(gsum, Wh, bh, out);
}